// SNN_Vanilla_11630771437863
// MI455X (gfx1250) — compile-verified
//
#include <hip/hip_runtime.h>

typedef float v2f __attribute__((ext_vector_type(2)));
typedef float v8f __attribute__((ext_vector_type(8)));

#define T_STEPS 100
#define BATCH   256
#define DIM     2312          // 2*34*34
#define HID     1000
#define OUTN    10
#define MTOT    (T_STEPS*BATCH)   // 25600

#define BM 128
#define BN 64
#define BK 32
#define ASTR 36
#define BSTR 36

// ---------------------------------------------------------------------------
// Phase 1: cur1_all[t*B+b, h] = sum_d data[b,t,d]*w1[h,d] + b1[h]
// fp32 WMMA 16x16x4, 8 waves/WG, wave tile 32x32, WG tile 128x64.
// grid.x = N tiles (16)  -> keeps the 1.2MB A slab hot in L2 across N sweep
// grid.y = M tiles (200)
// ---------------------------------------------------------------------------
__global__ __launch_bounds__(256) void snn_gemm1_kernel(
    const float* __restrict__ data,   // [B, T, D]
    const float* __restrict__ w1,     // [H, D]
    const float* __restrict__ b1,     // [H]
    float* __restrict__ cur1)         // [T*B, H]
{
  __shared__ __align__(16) float As[BM][ASTR];
  __shared__ __align__(16) float Bs[BN][BSTR];

  const int tid   = threadIdx.x;
  const int lane  = tid & 31;
  const int wave  = tid >> 5;       // 0..7
  const int wm    = wave >> 1;      // 0..3 (M)
  const int wn    = wave & 1;       // 0..1 (N)
  const int lmod  = lane & 15;
  const int lhalf = lane >> 4;      // 0/1

  const int n0 = blockIdx.x * BN;
  const int m0 = blockIdx.y * BM;

  v8f acc00 = {}; v8f acc01 = {}; v8f acc10 = {}; v8f acc11 = {};

  const int ldrow = tid >> 3;           // 0..31
  const int kq    = (tid & 7) * 4;      // 0,4,...,28

  for (int k0 = 0; k0 < DIM; k0 += BK) {
    // Stage A: 128 rows x 32 k (row m -> data[b= m&255, t= m>>8, :])
#pragma unroll
    for (int i = 0; i < 4; ++i) {
      int row = ldrow + 32 * i;
      int m   = m0 + row;
      int t   = m >> 8;
      int b   = m & 255;
      int k   = k0 + kq;
      float4 v = make_float4(0.f, 0.f, 0.f, 0.f);
      if (k < DIM)   // DIM%4==0 and kq%4==0 -> float4 fully valid or fully out
        v = *(const float4*)(data + (size_t)b * (T_STEPS * DIM) + (size_t)t * DIM + k);
      *(float4*)&As[row][kq] = v;
    }
    // Stage B (w1 transposed): 64 n-rows x 32 k
#pragma unroll
    for (int i = 0; i < 2; ++i) {
      int n = n0 + ldrow + 32 * i;
      int k = k0 + kq;
      float4 v = make_float4(0.f, 0.f, 0.f, 0.f);
      if (n < HID && k < DIM)
        v = *(const float4*)(w1 + (size_t)n * DIM + k);
      *(float4*)&Bs[ldrow + 32 * i][kq] = v;
    }
    __syncthreads();

    // 8 x (4 WMMA f32 16x16x4)
#pragma unroll
    for (int kk = 0; kk < BK; kk += 4) {
      int kb = kk + 2 * lhalf;   // even -> 8B aligned LDS reads
      v2f a0 = *(const v2f*)&As[wm * 32 + lmod][kb];
      v2f a1 = *(const v2f*)&As[wm * 32 + 16 + lmod][kb];
      v2f b0 = *(const v2f*)&Bs[wn * 32 + lmod][kb];
      v2f b1v = *(const v2f*)&Bs[wn * 32 + 16 + lmod][kb];
      acc00 = __builtin_amdgcn_wmma_f32_16x16x4_f32(false, a0, false, b0,  (short)0, acc00, false, false);
      acc01 = __builtin_amdgcn_wmma_f32_16x16x4_f32(false, a0, false, b1v, (short)0, acc01, false, false);
      acc10 = __builtin_amdgcn_wmma_f32_16x16x4_f32(false, a1, false, b0,  (short)0, acc10, false, false);
      acc11 = __builtin_amdgcn_wmma_f32_16x16x4_f32(false, a1, false, b1v, (short)0, acc11, false, false);
    }
    __syncthreads();
  }

  // Store C + bias. C layout: VGPR r -> M = r + 8*lhalf, N = lmod.
  const int gnA = n0 + wn * 32 + lmod;
  const int gnB = gnA + 16;
#pragma unroll
  for (int r = 0; r < 8; ++r) {
    int gm = m0 + wm * 32 + r + 8 * lhalf;
    if (gnA < HID) {
      cur1[(size_t)gm        * HID + gnA] = acc00[r] + b1[gnA];
      cur1[(size_t)(gm + 16) * HID + gnA] = acc10[r] + b1[gnA];
    }
    if (gnB < HID) {
      cur1[(size_t)gm        * HID + gnB] = acc01[r] + b1[gnB];
      cur1[(size_t)(gm + 16) * HID + gnB] = acc11[r] + b1[gnB];
    }
  }
}

// ---------------------------------------------------------------------------
// Phase 2: one launch per step t in [0..T]:
//   blocks 0..999     : layer-1 leaky update for step t          (if t < T)
//   blocks 1000..1079 : layer-2 dot + leaky update for step t-1  (if t >= 1)
// spk1 is ping-pong buffered: this launch reads S[(t-1)&1], writes S[t&1],
// so the two halves never race.
// ---------------------------------------------------------------------------
__global__ __launch_bounds__(256) void snn_step_kernel(
    const float* __restrict__ cur1,   // [T*B, H]
    const float* __restrict__ w2,     // [O, H]
    const float* __restrict__ b2,     // [O]
    const float* __restrict__ beta_p,
    float* __restrict__ mem1,         // [B*H]
    const float* __restrict__ spk1_prev,
    float* __restrict__ spk1_cur,
    float* __restrict__ mem2,         // [B*O]
    float* __restrict__ spk2,         // [B*O]
    float* __restrict__ out_spk,      // [T,B,O]
    float* __restrict__ out_mem,      // [T,B,O]
    int t)
{
  const float beta = *beta_p;
  const int bx = blockIdx.x;

  if (bx < (BATCH * HID) / 256) {
    if (t >= T_STEPS) return;
    int e = bx * 256 + threadIdx.x;            // 0..255999  (= b*H + h)
    float m = beta * mem1[e] + cur1[(size_t)t * (BATCH * HID) + e] - spk1_prev[e];
    float s = (m - 1.0f) > 0.0f ? 1.0f : 0.0f;
    mem1[e] = m;
    spk1_cur[e] = s;
  } else {
    if (t < 1) return;
    const int tp   = t - 1;
    const int lane = threadIdx.x & 31;
    const int wid  = (bx - (BATCH * HID) / 256) * 8 + (threadIdx.x >> 5); // 0..639
#pragma unroll
    for (int i = 0; i < 4; ++i) {
      int p = wid * 4 + i;                     // 0..2559
      int b = p / OUTN, o = p - b * OUTN;
      float sum = 0.f;
      for (int h = lane; h < HID; h += 32)
        sum += spk1_prev[b * HID + h] * w2[o * HID + h];
#pragma unroll
      for (int off = 16; off > 0; off >>= 1)
        sum += __shfl_xor(sum, off, 32);
      if (lane == 0) {
        int idx = b * OUTN + o;
        float m2 = beta * mem2[idx] + (sum + b2[o]) - spk2[idx];
        float s2 = (m2 - 1.0f) > 0.0f ? 1.0f : 0.0f;
        mem2[idx] = m2;
        spk2[idx] = s2;
        out_spk[(size_t)(tp * BATCH + b) * OUTN + o] = s2;
        out_mem[(size_t)(tp * BATCH + b) * OUTN + o] = m2;
      }
    }
  }
}

// ---------------------------------------------------------------------------
extern "C" void kernel_launch(void* const* d_in, const int* in_sizes, int n_in,
                              void* d_out, int out_size, void* d_ws, size_t ws_size,
                              hipStream_t stream) {
  (void)in_sizes; (void)n_in; (void)out_size; (void)ws_size;
  const float* data = (const float*)d_in[0];
  const float* w1   = (const float*)d_in[1];
  const float* b1   = (const float*)d_in[2];
  const float* w2   = (const float*)d_in[3];
  const float* b2   = (const float*)d_in[4];
  const float* beta = (const float*)d_in[5];
  float* out = (float*)d_out;

  // Workspace layout (floats):
  float* cur1 = (float*)d_ws;                          // 25,600,000
  float* mem1 = cur1 + (size_t)MTOT * HID;             // 256,000
  float* s0   = mem1 + (size_t)BATCH * HID;            // 256,000
  float* s1   = s0   + (size_t)BATCH * HID;            // 256,000
  float* mem2 = s1   + (size_t)BATCH * HID;            // 2,560
  float* spk2 = mem2 + (size_t)BATCH * OUTN;           // 2,560

  // Zero all recurrent state (capture-safe memset node).
  hipMemsetAsync(mem1, 0,
                 (3 * (size_t)BATCH * HID + 2 * (size_t)BATCH * OUTN) * sizeof(float),
                 stream);

  dim3 g((HID + BN - 1) / BN, MTOT / BM);              // (16, 200)
  snn_gemm1_kernel<<<g, 256, 0, stream>>>(data, w1, b1, cur1);

  float* out_spk = out;
  float* out_mem = out + (size_t)T_STEPS * BATCH * OUTN;

  for (int t = 0; t <= T_STEPS; ++t) {
    float* prev = (t & 1) ? s0 : s1;   // S[(t-1)&1]
    float* cur  = (t & 1) ? s1 : s0;   // S[t&1]
    snn_step_kernel<<<1080, 256, 0, stream>>>(cur1, w2, b2, beta, mem1, prev, cur,
                                              mem2, spk2, out_spk, out_mem, t);
  }
}